// MultiLabelEmbeddingLayer_4827543241412
// MI455X (gfx1250) — compile-verified
//
#include <hip/hip_runtime.h>
#include <cstdint>

// Multi-label embedding gather-sum for MI455X (gfx1250).
// out[token, :] = sum_k W[labels[token,k], :],  D=128 fp32, K=4.
//
// One wave32 per token-slot, 4 tokens per wave, double-buffered CDNA5 async
// DMA (global_load_async_to_lds_b128, GVS saddr form, tracked by ASYNCcnt):
// while the current token's 4 rows are reduced from LDS, the next token's 4
// row-DMAs are already in flight. Output uses non-temporal stores so the
// 67MB write stream does not evict the L2-resident 25.6MB embedding table.

#define D_DIM   128
#define K_LAB   4
#define WAVES_PER_BLOCK 8
#define TOK_PER_WAVE 4
#define ROW_BYTES (D_DIM * 4)   // 512B per embedding row

typedef float v4f __attribute__((ext_vector_type(4)));

__device__ __forceinline__ void issue_row_dmas(const int* __restrict__ labels,
                                               uint64_t wbase_sgpr,
                                               int token, int lane,
                                               uint32_t lds_lane_base)
{
    // 4 labels for this token (wave-uniform 16B load).
    const int4 lab = reinterpret_cast<const int4*>(labels)[token];

    // 32-bit per-lane offsets into W: label*512 + lane*16 (max 25.6MB < 2^31).
    const uint32_t lo = (uint32_t)lane * 16u;
    const uint32_t o0 = (uint32_t)lab.x * ROW_BYTES + lo;
    const uint32_t o1 = (uint32_t)lab.y * ROW_BYTES + lo;
    const uint32_t o2 = (uint32_t)lab.z * ROW_BYTES + lo;
    const uint32_t o3 = (uint32_t)lab.w * ROW_BYTES + lo;

    // GVS form: 64-bit SGPR base + 32-bit VGPR offset. Each instruction DMAs
    // a full 512B row (32 lanes x 16B) from global memory into LDS.
    asm volatile("global_load_async_to_lds_b128 %0, %1, %2"
                 :: "v"(lds_lane_base + 0u * ROW_BYTES), "v"(o0), "s"(wbase_sgpr) : "memory");
    asm volatile("global_load_async_to_lds_b128 %0, %1, %2"
                 :: "v"(lds_lane_base + 1u * ROW_BYTES), "v"(o1), "s"(wbase_sgpr) : "memory");
    asm volatile("global_load_async_to_lds_b128 %0, %1, %2"
                 :: "v"(lds_lane_base + 2u * ROW_BYTES), "v"(o2), "s"(wbase_sgpr) : "memory");
    asm volatile("global_load_async_to_lds_b128 %0, %1, %2"
                 :: "v"(lds_lane_base + 3u * ROW_BYTES), "v"(o3), "s"(wbase_sgpr) : "memory");
}

__global__ __launch_bounds__(WAVES_PER_BLOCK * 32, 1)
void embed_sum_async_kernel(const int* __restrict__ labels,
                            const float* __restrict__ W,
                            float* __restrict__ out,
                            int n_tokens)
{
    // Per-wave double buffer: 2 x (4 rows x 512B) = 4KB; 8 waves -> 32KB LDS.
    __shared__ float smem[WAVES_PER_BLOCK][2][K_LAB * D_DIM];

    const int lane = threadIdx.x & 31;
    const int wave = threadIdx.x >> 5;

    // Wave w handles tokens base+w, base+w+8, ... so each iteration the 8
    // waves of a block touch contiguous labels (128B) and output (4KB).
    const int base = blockIdx.x * (WAVES_PER_BLOCK * TOK_PER_WAVE) + wave;

    // Low 32 bits of a generic LDS pointer == LDS byte address
    // (shared aperture lives in ADDR[63:32] per the CDNA5 aperture rules).
    const uint32_t lane_off = (uint32_t)lane * 16u;
    const uint32_t lds_buf[2] = {
        (uint32_t)(uintptr_t)&smem[wave][0][0] + lane_off,
        (uint32_t)(uintptr_t)&smem[wave][1][0] + lane_off
    };

    const uint64_t wbase = (uint64_t)(uintptr_t)W;

    if (base >= n_tokens) return;   // wave-uniform: EXEC stays all-1s

    // Prologue: stage token 0 into buffer 0.
    issue_row_dmas(labels, wbase, base, lane, lds_buf[0]);

#pragma unroll
    for (int i = 0; i < TOK_PER_WAVE; ++i) {
        const int tcur = base + i * WAVES_PER_BLOCK;
        if (tcur >= n_tokens) break;

        const int tnext = tcur + WAVES_PER_BLOCK;
        const bool have_next = (i + 1 < TOK_PER_WAVE) && (tnext < n_tokens);

        if (have_next) {
            // Stage next token into the other buffer, then wait until only
            // those 4 new DMAs remain outstanding: async ops complete in
            // order, so ASYNCcnt <= 4 means the current buffer has landed.
            issue_row_dmas(labels, wbase, tnext, lane, lds_buf[(i + 1) & 1]);
            asm volatile("s_wait_asynccnt 0x4" ::: "memory");
        } else {
            asm volatile("s_wait_asynccnt 0x0" ::: "memory");
        }

        // Reduce the 4 staged rows of the current buffer (ds_load_b128 each).
        const v4f* s4 = reinterpret_cast<const v4f*>(&smem[wave][i & 1][0]);
        v4f a = s4[0 * 32 + lane];
        v4f b = s4[1 * 32 + lane];
        v4f c = s4[2 * 32 + lane];
        v4f d = s4[3 * 32 + lane];
        v4f r = (a + b) + (c + d);

        // Non-temporal streaming store: keep W resident in the 192MB L2.
        v4f* o = reinterpret_cast<v4f*>(out + (size_t)tcur * D_DIM) + lane;
        __builtin_nontemporal_store(r, o);
    }
}

extern "C" void kernel_launch(void* const* d_in, const int* in_sizes, int n_in,
                              void* d_out, int out_size, void* d_ws, size_t ws_size,
                              hipStream_t stream) {
    const int*   labels = (const int*)d_in[0];   // [B, S, K] int32
    const float* W      = (const float*)d_in[1]; // [VOCAB, D] fp32
    float*       out    = (float*)d_out;         // [B, S, D] fp32

    const int n_tokens = in_sizes[0] / K_LAB;    // B*S = 131072
    const int tokens_per_block = WAVES_PER_BLOCK * TOK_PER_WAVE;
    dim3 block(WAVES_PER_BLOCK * 32);
    dim3 grid((n_tokens + tokens_per_block - 1) / tokens_per_block);
    embed_sum_async_kernel<<<grid, block, 0, stream>>>(labels, W, out, n_tokens);
}